// QuantumLayer_80427557585182
// MI455X (gfx1250) — compile-verified
//
#include <hip/hip_runtime.h>

typedef __attribute__((ext_vector_type(2))) float v2f;
typedef __attribute__((ext_vector_type(8))) float v8f;

#define QDIM 16  // 2^4 amplitudes per sample

// One wave handles a tile of 16 samples:
//   B operand = normalized sample tile Xn^T  (K=j amplitude index, N=sample)
//   A operand = Ur / Ui rows                 (M=k output basis, K=j)
//   D = U * Xn^T  -> lane holds column (=sample), VGPR+half holds k
// probs = Dr^2 + Di^2 ; y[i] = sum_k probs[k] * (1-2*bit_{3-i}(k))
__global__ __launch_bounds__(256) void quantum_layer_wmma(
    const float* __restrict__ x, const float* __restrict__ qw,
    float* __restrict__ out, int nTiles)
{
  const int lane = threadIdx.x & 31;
  const int grp  = lane >> 4;    // which 16-lane half
  const int col  = lane & 15;    // A: row k ; B/D: sample within tile

  // cos/sin of half-angles via HW transcendentals (v_cos_f32/v_sin_f32 take
  // revolutions: scale radians by 1/2pi). Runs once per wave.
  const float INV2PI = 0.15915494309189535f;
  float cc[4], ss[4];
#pragma unroll
  for (int i = 0; i < 4; ++i) {
    const float r = 0.5f * qw[i] * INV2PI;
    cc[i] = __builtin_amdgcn_cosf(r);
    ss[i] = __builtin_amdgcn_sinf(r);
  }

  // Build A operands once: U[k][j] = (-i)^popcount(k^j) * prod_p (bits equal ? c : s)
  // A layout (f32 16x4 chunk q): lane l, vgpr v -> M=l%16, K = 4q + 2*(l/16) + v
  v2f ar[4], ai[4];
#pragma unroll
  for (int q = 0; q < 4; ++q) {
#pragma unroll
    for (int v = 0; v < 2; ++v) {
      const int j = 4 * q + 2 * grp + v;
      float m = 1.0f;
      int d = 0;
#pragma unroll
      for (int p = 0; p < 4; ++p) {           // bit position p (0 = LSB) <-> qubit 3-p
        const int bk = (col >> p) & 1;
        const int bj = (j   >> p) & 1;
        m *= (bk == bj) ? cc[3 - p] : ss[3 - p];
        d += bk ^ bj;
      }
      d &= 3;
      const float ur = (d == 0) ? m : ((d == 2) ? -m : 0.0f);
      const float ui = (d == 1) ? -m : ((d == 3) ? m : 0.0f);
      if (v == 0) { ar[q].x = ur; ai[q].x = ui; }
      else        { ar[q].y = ur; ai[q].y = ui; }
    }
  }

  const int wave   = (int)((blockIdx.x * blockDim.x + threadIdx.x) >> 5);
  const int nWaves = (int)((gridDim.x * blockDim.x) >> 5);
  const int j0     = 2 * grp;

  for (int t = wave; t < nTiles; t += nWaves) {
    const float* row = x + ((size_t)t * 16 + col) * QDIM;
    if (t + nWaves < nTiles)  // pull next tile toward the caches (global_prefetch_b8)
      __builtin_prefetch(x + ((size_t)(t + nWaves) * 16 + col) * QDIM, 0, 0);

    // B layout (f32 4x16 chunk q): lane l, vgpr v -> K = 4q + 2*(l/16) + v, N = l%16
    v2f b[4];
#pragma unroll
    for (int q = 0; q < 4; ++q)
      b[q] = *(const v2f*)(row + 4 * q + j0);

    // NaN scrub + per-sample L2 norm (each sample split across a lane pair)
    float ps = 0.0f;
#pragma unroll
    for (int q = 0; q < 4; ++q) {
      float bx = b[q].x, by = b[q].y;
      bx = (bx != bx) ? 0.0f : bx;
      by = (by != by) ? 0.0f : by;
      b[q].x = bx; b[q].y = by;
      ps += bx * bx + by * by;
    }
    const float tot = ps + __shfl_xor(ps, 16, 32);
    const float inv = __builtin_amdgcn_rsqf(tot);   // v_rsq_f32
#pragma unroll
    for (int q = 0; q < 4; ++q) { b[q].x *= inv; b[q].y *= inv; }

    // Real & imag 16x16x16 matmuls as 4 chained K=4 f32 WMMAs each
    v8f cr = {};
    v8f ci = {};
#pragma unroll
    for (int q = 0; q < 4; ++q) {
      cr = __builtin_amdgcn_wmma_f32_16x16x4_f32(false, ar[q], false, b[q],
                                                 (short)0, cr, false, false);
      ci = __builtin_amdgcn_wmma_f32_16x16x4_f32(false, ai[q], false, b[q],
                                                 (short)0, ci, false, false);
    }

    // probs: element i2 in this lane is basis k = i2 + 8*grp, sample = col
    float p[8];
#pragma unroll
    for (int i2 = 0; i2 < 8; ++i2) p[i2] = cr[i2] * cr[i2] + ci[i2] * ci[i2];

    // Signed sums over k: qubit i uses bit (3-i) of k. bits 0..2 are bits of the
    // VGPR index (compile-time signs); bit 3 == grp; the other 8 k's live in lane^16.
    const float a01 = p[0] + p[1], a23 = p[2] + p[3], a45 = p[4] + p[5], a67 = p[6] + p[7];
    const float d01 = p[0] - p[1], d23 = p[2] - p[3], d45 = p[4] - p[5], d67 = p[6] - p[7];
    const float q03 = a01 + a23, q47 = a45 + a67;
    float y3 = (d01 + d23) + (d45 + d67);           // qubit 3: bit0 of k
    float y2 = (a01 - a23) + (a45 - a67);           // qubit 2: bit1 of k
    float y1 = q03 - q47;                           // qubit 1: bit2 of k
    float y0 = grp ? -(q03 + q47) : (q03 + q47);    // qubit 0: bit3 of k (== grp)

    y0 += __shfl_xor(y0, 16, 32);
    y1 += __shfl_xor(y1, 16, 32);
    y2 += __shfl_xor(y2, 16, 32);
    y3 += __shfl_xor(y3, 16, 32);

    if (grp == 0) {  // lanes 0..15 each own one sample -> contiguous 16B store
      float4 o; o.x = y0; o.y = y1; o.z = y2; o.w = y3;
      *(float4*)(out + ((size_t)t * 16 + col) * 4) = o;
    }
  }
}

extern "C" void kernel_launch(void* const* d_in, const int* in_sizes, int n_in,
                              void* d_out, int out_size, void* d_ws, size_t ws_size,
                              hipStream_t stream) {
  const float* x  = (const float*)d_in[0];   // [BATCH, 16] f32
  const float* qw = (const float*)d_in[1];   // [4] f32
  float* out = (float*)d_out;                // [BATCH, 4] f32

  const int nTiles = in_sizes[0] / (QDIM * 16);  // 16 samples per tile
  int blocks = 2048;                              // 8 waves/block -> 16384 waves
  const int wavesPerBlock = 8;
  if (blocks * wavesPerBlock > nTiles) blocks = (nTiles + wavesPerBlock - 1) / wavesPerBlock;
  if (blocks < 1) blocks = 1;

  quantum_layer_wmma<<<blocks, 256, 0, stream>>>(x, qw, out, nTiles);
}